// MultiSimilarityLoss_19731079758542
// MI455X (gfx1250) — compile-verified
//
#include <hip/hip_runtime.h>
#include <hip/hip_bf16.h>
#include <math.h>

// CDNA5 / gfx1250 WMMA vector types
typedef float v2f __attribute__((ext_vector_type(2)));
typedef float v8f __attribute__((ext_vector_type(8)));

#define BSZ     8192
#define DDIM    128
#define ALPHA   2.0f
#define BETA    50.0f
#define BASEC   0.5f
#define MARGINC 0.1f
#define NTILES  (BSZ / 16)   // 512 column tiles of 16
#define KCHUNKS (DDIM / 4)   // 32 k-steps of 4 for V_WMMA_F32_16X16X4_F32

// --- wave32 butterfly reductions within each 16-lane half (masks 1,2,4,8 only
// flip the low 4 lane bits, so lanes 0-15 and 16-31 reduce independently) ---
__device__ __forceinline__ float red_min16(float v) {
    v = fminf(v, __shfl_xor(v, 1));
    v = fminf(v, __shfl_xor(v, 2));
    v = fminf(v, __shfl_xor(v, 4));
    v = fminf(v, __shfl_xor(v, 8));
    return v;
}
__device__ __forceinline__ float red_max16(float v) {
    v = fmaxf(v, __shfl_xor(v, 1));
    v = fmaxf(v, __shfl_xor(v, 2));
    v = fmaxf(v, __shfl_xor(v, 4));
    v = fmaxf(v, __shfl_xor(v, 8));
    return v;
}
__device__ __forceinline__ float red_sum16(float v) {
    v += __shfl_xor(v, 1);
    v += __shfl_xor(v, 2);
    v += __shfl_xor(v, 4);
    v += __shfl_xor(v, 8);
    return v;
}

// One block per 16-row tile. 8 waves split the 512 column tiles.
// 3 passes over columns (recompute sim via f32 WMMA each pass; E is 4MB ->
// permanently L2-resident, so recompute beats materializing 268MB of sims).
__global__ __launch_bounds__(256) void ms_loss_main(
    const float* __restrict__ emb,     // [8192][128], L2-normalized
    const int*   __restrict__ labels,  // [8192]
    float*       __restrict__ partial) // [512][2] : (loss_sum, valid_count)
{
    __shared__ float ldsA[8][16];
    __shared__ float ldsB[8][16];
    __shared__ float row_pmin[16], row_nany[16];
    __shared__ float row_nmax[16], row_nsum[16];
    __shared__ float row_psum[16], row_pcnt[16];

    const int i0   = blockIdx.x * 16;
    const int tid  = threadIdx.x;
    const int wave = tid >> 5;
    const int lane = tid & 31;
    const int lo   = lane & 15;     // column index within tile (C-layout N)
    const int hi   = lane >> 4;     // selects rows r+8*hi (C-layout M half)

    // Preload all 32 A-fragments for this row tile (resident: 64 VGPRs).
    // f32 16x16x4 A layout: lanes 0-15 hold M=lane, K = {0,1}; lanes 16-31 hold K = {2,3}.
    v2f afrag[KCHUNKS];
    {
        const float* arow = emb + (size_t)(i0 + lo) * DDIM + 2 * hi;
#pragma unroll
        for (int k = 0; k < KCHUNKS; ++k)
            afrag[k] = *(const v2f*)(arow + 4 * k);
    }

    // Row labels for the 8 rows this lane covers (M = r + 8*hi)
    int rl[8];
#pragma unroll
    for (int r = 0; r < 8; ++r) rl[r] = labels[i0 + r + 8 * hi];

    // ---------------- pass 1: pos_min (excl. self), has_neg ----------------
    float pmin[8], nany[8];
#pragma unroll
    for (int r = 0; r < 8; ++r) { pmin[r] = INFINITY; nany[r] = 0.0f; }

    for (int jt = wave; jt < NTILES; jt += 8) {
        const int j0 = jt * 16;
        const float* brow = emb + (size_t)(j0 + lo) * DDIM + 2 * hi;
        const int lc = labels[j0 + lo];
        v8f c = {};
#pragma unroll
        for (int k = 0; k < KCHUNKS; ++k) {
            v2f b = *(const v2f*)(brow + 4 * k);
            c = __builtin_amdgcn_wmma_f32_16x16x4_f32(
                    false, afrag[k], false, b, (short)0, c, false, false);
        }
#pragma unroll
        for (int r = 0; r < 8; ++r) {
            const float s = c[r];
            const bool same = (rl[r] == lc);
            const bool self = ((i0 + r + 8 * hi) == (j0 + lo));
            if (same && !self) pmin[r] = fminf(pmin[r], s);
            if (!same)         nany[r] = 1.0f;
        }
    }
#pragma unroll
    for (int r = 0; r < 8; ++r) { pmin[r] = red_min16(pmin[r]); nany[r] = red_max16(nany[r]); }
    if (lo == 0) {
#pragma unroll
        for (int r = 0; r < 8; ++r) { ldsA[wave][r + 8 * hi] = pmin[r]; ldsB[wave][r + 8 * hi] = nany[r]; }
    }
    __syncthreads();
    if (tid < 16) {
        float m = ldsA[0][tid], ny = ldsB[0][tid];
        for (int w = 1; w < 8; ++w) { m = fminf(m, ldsA[w][tid]); ny = fmaxf(ny, ldsB[w][tid]); }
        row_pmin[tid] = m; row_nany[tid] = ny;
    }
    __syncthreads();

    float pth[8];  // neg_keep threshold: sim > pos_min - margin
#pragma unroll
    for (int r = 0; r < 8; ++r) pth[r] = row_pmin[r + 8 * hi] - MARGINC;

    // ------------- pass 2: neg_max over kept negs + neg_sum -------------
    float nmax[8], nsum[8];
#pragma unroll
    for (int r = 0; r < 8; ++r) { nmax[r] = -INFINITY; nsum[r] = 0.0f; }

    for (int jt = wave; jt < NTILES; jt += 8) {
        const int j0 = jt * 16;
        const float* brow = emb + (size_t)(j0 + lo) * DDIM + 2 * hi;
        const int lc = labels[j0 + lo];
        v8f c = {};
#pragma unroll
        for (int k = 0; k < KCHUNKS; ++k) {
            v2f b = *(const v2f*)(brow + 4 * k);
            c = __builtin_amdgcn_wmma_f32_16x16x4_f32(
                    false, afrag[k], false, b, (short)0, c, false, false);
        }
#pragma unroll
        for (int r = 0; r < 8; ++r) {
            const float s = c[r];
            const bool same = (rl[r] == lc);
            if (!same && (s > pth[r])) {
                nmax[r] = fmaxf(nmax[r], s);
                nsum[r] += expf(BETA * (s - BASEC));
            }
        }
    }
#pragma unroll
    for (int r = 0; r < 8; ++r) { nmax[r] = red_max16(nmax[r]); nsum[r] = red_sum16(nsum[r]); }
    if (lo == 0) {
#pragma unroll
        for (int r = 0; r < 8; ++r) { ldsA[wave][r + 8 * hi] = nmax[r]; ldsB[wave][r + 8 * hi] = nsum[r]; }
    }
    __syncthreads();
    if (tid < 16) {
        float m = ldsA[0][tid], s = ldsB[0][tid];
        for (int w = 1; w < 8; ++w) { m = fmaxf(m, ldsA[w][tid]); s += ldsB[w][tid]; }
        row_nmax[tid] = m; row_nsum[tid] = s;
    }
    __syncthreads();

    float nth[8];  // pos_keep threshold: sim < neg_max + margin
#pragma unroll
    for (int r = 0; r < 8; ++r) nth[r] = row_nmax[r + 8 * hi] + MARGINC;

    // ------------- pass 3: pos_sum over kept positives + count -------------
    float psum[8], pcnt[8];
#pragma unroll
    for (int r = 0; r < 8; ++r) { psum[r] = 0.0f; pcnt[r] = 0.0f; }

    for (int jt = wave; jt < NTILES; jt += 8) {
        const int j0 = jt * 16;
        const float* brow = emb + (size_t)(j0 + lo) * DDIM + 2 * hi;
        const int lc = labels[j0 + lo];
        v8f c = {};
#pragma unroll
        for (int k = 0; k < KCHUNKS; ++k) {
            v2f b = *(const v2f*)(brow + 4 * k);
            c = __builtin_amdgcn_wmma_f32_16x16x4_f32(
                    false, afrag[k], false, b, (short)0, c, false, false);
        }
#pragma unroll
        for (int r = 0; r < 8; ++r) {
            const float s = c[r];
            const bool same = (rl[r] == lc);
            const bool self = ((i0 + r + 8 * hi) == (j0 + lo));
            if (same && !self && (s < nth[r])) {
                psum[r] += expf(-ALPHA * (s - BASEC));
                pcnt[r] += 1.0f;
            }
        }
    }
#pragma unroll
    for (int r = 0; r < 8; ++r) { psum[r] = red_sum16(psum[r]); pcnt[r] = red_sum16(pcnt[r]); }
    if (lo == 0) {
#pragma unroll
        for (int r = 0; r < 8; ++r) { ldsA[wave][r + 8 * hi] = psum[r]; ldsB[wave][r + 8 * hi] = pcnt[r]; }
    }
    __syncthreads();
    if (tid < 16) {
        float s = ldsA[0][tid], n = ldsB[0][tid];
        for (int w = 1; w < 8; ++w) { s += ldsA[w][tid]; n += ldsB[w][tid]; }
        row_psum[tid] = s; row_pcnt[tid] = n;
    }
    __syncthreads();

    // ------------- per-row loss, deterministic block partial -------------
    if (tid < 16) {
        const bool has_pos = (row_pmin[tid] < INFINITY);
        const bool has_neg = (row_nany[tid] > 0.5f);
        const bool has_nk  = (row_nmax[tid] > -INFINITY);
        const bool has_pk  = (row_pcnt[tid] > 0.5f);
        const bool valid   = has_pos && has_neg && has_nk && has_pk;

        const float loss = log1pf(row_psum[tid]) / ALPHA + log1pf(row_nsum[tid]) / BETA;
        float tsum = valid ? loss : 0.0f;
        float tcnt = valid ? 1.0f : 0.0f;
        tsum = red_sum16(tsum);           // tid<16 are lanes 0-15 of wave 0
        tcnt = red_sum16(tcnt);
        if (tid == 0) {
            partial[2 * blockIdx.x + 0] = tsum;
            partial[2 * blockIdx.x + 1] = tcnt;
        }
    }
}

// Single-block deterministic finalize: mean over valid rows.
__global__ __launch_bounds__(256) void ms_loss_final(
    const float* __restrict__ partial, float* __restrict__ out)
{
    __shared__ float s_sum[256];
    __shared__ float s_cnt[256];
    float ts = 0.0f, tc = 0.0f;
    for (int i = threadIdx.x; i < (BSZ / 16); i += 256) {
        ts += partial[2 * i + 0];
        tc += partial[2 * i + 1];
    }
    s_sum[threadIdx.x] = ts;
    s_cnt[threadIdx.x] = tc;
    __syncthreads();
    for (int off = 128; off > 0; off >>= 1) {
        if (threadIdx.x < off) {
            s_sum[threadIdx.x] += s_sum[threadIdx.x + off];
            s_cnt[threadIdx.x] += s_cnt[threadIdx.x + off];
        }
        __syncthreads();
    }
    if (threadIdx.x == 0) {
        const float total = s_sum[0];
        const float count = s_cnt[0];
        out[0] = (count > 0.0f) ? (total / fmaxf(count, 1.0f)) : 0.0f;
    }
}

extern "C" void kernel_launch(void* const* d_in, const int* in_sizes, int n_in,
                              void* d_out, int out_size, void* d_ws, size_t ws_size,
                              hipStream_t stream) {
    const float* emb    = (const float*)d_in[0];   // [8192*128] f32
    const int*   labels = (const int*)d_in[1];     // [8192] i32
    float*       out    = (float*)d_out;           // scalar f32
    float*       part   = (float*)d_ws;            // 512*2 f32 = 4KB scratch

    ms_loss_main<<<BSZ / 16, 256, 0, stream>>>(emb, labels, part);
    ms_loss_final<<<1, 256, 0, stream>>>(part, out);
}